// SymmetricContraction_85779086836276
// MI455X (gfx1250) — compile-verified
//
#include <hip/hip_runtime.h>

#ifndef __has_builtin
#define __has_builtin(x) 0
#endif

#define MUL    128
#define IRREP  9
#define NPATH  9
#define NELEM  64
#define ROWS   8                       // batch rows per block iteration
#define TPB    (MUL * ROWS)            // 1024 threads = 32 wave32
#define W_ELEMS (NELEM * NPATH * MUL)  // 73728 floats = 294912 bytes LDS

typedef __attribute__((ext_vector_type(4))) unsigned int tdm_v4u;
typedef __attribute__((ext_vector_type(8))) int          tdm_v8i;
typedef __attribute__((ext_vector_type(4))) int          tdm_v4i;

__global__ __launch_bounds__(TPB, 1)
void SymmetricContraction_85779086836276_kernel(const float* __restrict__ x,
                                                const int*   __restrict__ indices,
                                                const float* __restrict__ weight,
                                                float*       __restrict__ out,
                                                int batch)
{
    extern __shared__ float lds_w[];   // NELEM x NPATH x MUL fp32 (288 KB)
    const int tid = (int)threadIdx.x;

    // ---------------- Stage full weight tensor into LDS ----------------
#if defined(__gfx1250__) && __has_builtin(__builtin_amdgcn_tensor_load_to_lds) && __has_builtin(__builtin_amdgcn_s_wait_tensorcnt)
    // TDM descriptor: 2D tile = whole tensor (1152 floats x 64 rows, stride 1152)
    // => contiguous 294912-byte DMA into LDS offset 0.
    // Wave-0 only; readfirstlane makes the branch scalar (TDM ignores EXEC,
    // so other waves must actually branch around the instruction).
    if (__builtin_amdgcn_readfirstlane(tid) < 32) {
        unsigned long long ga = (unsigned long long)(const void*)weight;
        tdm_v4u g0;
        g0.x = 1u;                                          // count = 1 (valid D#)
        g0.y = 0u;                                          // lds_addr = 0
        g0.z = (unsigned)(ga & 0xffffffffull);              // global_addr[31:0]
        g0.w = (unsigned)((ga >> 32) & 0x01ffffffull)       // global_addr[56:32]
             | 0x80000000u;                                 // type = 2 ("image")
        tdm_v8i g1;
        g1[0] = 0x00020000;                                 // data_size = 4B; mask=0
        g1[1] = (int)(1152u << 16);                         // tensor_dim0 = 1152 (lo16)
        g1[2] = (int)(64u   << 16);                         // tdim0 hi=0 | tensor_dim1 = 64
        g1[3] = (int)(1152u << 16);                         // tdim1 hi=0 | tile_dim0 = 1152
        g1[4] = 64;                                         // tile_dim1 = 64, tile_dim2 = 0
        g1[5] = 1152;                                       // tensor_dim0_stride = 1152
        g1[6] = 0;
        g1[7] = 0;                                          // tensor_dim1_stride = 0
        tdm_v4i gz = (tdm_v4i){0, 0, 0, 0};
#if __clang_major__ >= 23
        tdm_v8i gz8 = (tdm_v8i){0, 0, 0, 0, 0, 0, 0, 0};
        __builtin_amdgcn_tensor_load_to_lds(g0, g1, gz, gz, gz8, 0);
#else
        __builtin_amdgcn_tensor_load_to_lds(g0, g1, gz, gz, 0);
#endif
        __builtin_amdgcn_s_wait_tensorcnt(0);
    }
    __syncthreads();
#else
    for (int i = tid; i < W_ELEMS; i += TPB) lds_w[i] = weight[i];
    __syncthreads();
#endif

    // ---------------- Streaming compute ----------------
    const int m    = tid & (MUL - 1);         // irrep channel 0..127
    const int row  = tid >> 7;                // 0..ROWS-1
    const long long bstep = (long long)gridDim.x * ROWS;

    constexpr float C2 = 0.70710678118654752f;   // 1/sqrt(2)
    constexpr float C6 = 0.40824829046386302f;   // 1/sqrt(6)

    for (long long b = (long long)blockIdx.x * ROWS + row; b < batch; b += bstep) {
        const float* xb = x + b * (long long)(MUL * IRREP) + (long long)m * IRREP;

        if (b + bstep < batch)                        // prefetch next grid-stride chunk
            __builtin_prefetch(xb + bstep * (MUL * IRREP), 0, 0);

        // 9 NT dword loads: x streams once, keep it out of L2.
        float s  = __builtin_nontemporal_load(xb + 0);
        float v0 = __builtin_nontemporal_load(xb + 1);
        float v1 = __builtin_nontemporal_load(xb + 2);
        float v2 = __builtin_nontemporal_load(xb + 3);
        float q0 = __builtin_nontemporal_load(xb + 4);
        float q1 = __builtin_nontemporal_load(xb + 5);
        float q2 = __builtin_nontemporal_load(xb + 6);
        float q3 = __builtin_nontemporal_load(xb + 7);
        float q4 = __builtin_nontemporal_load(xb + 8);

        const int elem = indices[b];

        // Q = sum_a q_a B_a (symmetric traceless):
        float Qa = -C6 * q2 + C2 * q4;     // Q00
        float Qb = -C6 * q2 - C2 * q4;     // Q11
        float Qc = 2.0f * C6 * q2;         // Q22
        float Qd = C2 * q0;                // Q01
        float Qe = C2 * q1;                // Q12
        float Qf = C2 * q3;                // Q02

        float vv = v0 * v0 + v1 * v1 + v2 * v2;
        float qq = q0 * q0 + q1 * q1 + q2 * q2 + q3 * q3 + q4 * q4;

        float vQv = Qa * v0 * v0 + Qb * v1 * v1 + Qc * v2 * v2
                  + 2.0f * (Qd * v0 * v1 + Qe * v1 * v2 + Qf * v0 * v2);

        float d2 = Qd * Qd, e2 = Qe * Qe, f2 = Qf * Qf;
        float trQ3 = Qa * Qa * Qa + Qb * Qb * Qb + Qc * Qc * Qc
                   + 3.0f * (Qa * (d2 + f2) + Qb * (d2 + e2) + Qc * (e2 + f2))
                   + 6.0f * Qd * Qe * Qf;

        float ss = s * s;

        // weight row for this element: LDS, stride MUL between paths.
        // lanes m=0..31 map to distinct banks -> conflict-free ds_load_b32.
        const float* wr = lds_w + (long long)elem * (NPATH * MUL) + m;

        float acc;
        acc  = s       * wr[0 * MUL];   // s
        acc += ss      * wr[1 * MUL];   // s^2
        acc += vv      * wr[2 * MUL];   // |v|^2
        acc += qq      * wr[3 * MUL];   // |q|^2
        acc += ss * s  * wr[4 * MUL];   // s^3
        acc += s  * vv * wr[5 * MUL];   // s|v|^2
        acc += s  * qq * wr[6 * MUL];   // s|q|^2
        acc += vQv     * wr[7 * MUL];   // vQv
        acc += trQ3    * wr[8 * MUL];   // tr Q^3

        __builtin_nontemporal_store(acc, out + b * MUL + m);  // out streams once
    }
}

extern "C" void kernel_launch(void* const* d_in, const int* in_sizes, int n_in,
                              void* d_out, int out_size, void* d_ws, size_t ws_size,
                              hipStream_t stream) {
    const float* x       = (const float*)d_in[0];
    const int*   indices = (const int*)d_in[1];
    const float* weight  = (const float*)d_in[2];
    float*       out     = (float*)d_out;

    const int batch = in_sizes[1];                 // indices count == BATCH

    int blocks = (batch + ROWS - 1) / ROWS;
    if (blocks > 512) blocks = 512;                // grid-stride; amortize LDS preload

    const size_t shmem = (size_t)W_ELEMS * sizeof(float);   // 294912 B (< 320 KB/WGP)

    hipLaunchKernelGGL(SymmetricContraction_85779086836276_kernel,
                       dim3(blocks), dim3(TPB), shmem, stream,
                       x, indices, weight, out, batch);
}